// KineticModel_60266981097872
// MI455X (gfx1250) — compile-verified
//
#include <hip/hip_runtime.h>

// Problem constants (from the reference setup).
#define NS    8192      // species
#define NR    16384     // reactions
#define NBAL  7680
#define NUNBAL 512

typedef __attribute__((ext_vector_type(2))) float        v2f;
typedef __attribute__((ext_vector_type(8))) float        v8f;
typedef __attribute__((ext_vector_type(4))) unsigned int v4u;
typedef __attribute__((ext_vector_type(8))) int          v8i;
typedef __attribute__((ext_vector_type(4))) int          v4i;

// ---------------------------------------------------------------------------
// TDM: issue a 2D tensor_load_to_lds of a (tile_d1 rows x tile_d0 dwords) f32
// tile starting at gaddr, row stride = stride_d0 elements, into LDS at
// lds_addr with +4-dword padding per row (pad codes per ISA: interval code
// n -> 2^n 8-byte units; amount code m -> m+1 dwords).
// D# layout per CDNA5 ISA ch.8 (group0 128b, group1 256b, groups2/3 unused).
// This toolchain exposes the 6-arg builtin:
//   (uint32x4 g0, int32x8 g1, int32x4 g2, int32x4 g3, int32x8 pad, i32 cpol)
// ---------------------------------------------------------------------------
__device__ __forceinline__ void tdm_load_2d(unsigned lds_addr, const float* gaddr,
                                            unsigned tile_d0, unsigned tile_d1,
                                            unsigned tensor_d0, unsigned tensor_d1,
                                            unsigned stride_d0,
                                            unsigned pad_interval_code,
                                            unsigned pad_amount_code) {
  unsigned long long ga = (unsigned long long)(uintptr_t)gaddr;
  v4u g0;
  g0.x = 1u;                                           // count=1 valid user D#
  g0.y = lds_addr;                                     // LDS byte address
  g0.z = (unsigned)(ga & 0xFFFFFFFFu);                 // global_addr[31:0]
  g0.w = (unsigned)((ga >> 32) & 0x1FFFFFFu) | (2u << 30);  // addr[56:32] | type=2
  v8i g1;
  g1[0] = (int)((2u << 16)                             // data_size = 4 bytes
              | (1u << 20)                             // pad_enable
              | (pad_interval_code << 22)
              | (pad_amount_code << 25));
  g1[1] = (int)((tensor_d0 & 0xFFFFu) << 16);          // [63:48] tensor_dim0 lo
  g1[2] = (int)((tensor_d0 >> 16) | ((tensor_d1 & 0xFFFFu) << 16));
  g1[3] = (int)((tensor_d1 >> 16) | (tile_d0 << 16));  // [127:112] tile_dim0
  g1[4] = (int)(tile_d1 & 0xFFFFu);                    // tile_dim1, tile_dim2=0
  g1[5] = (int)stride_d0;                              // tensor_dim0_stride[31:0]
  g1[6] = 0;
  g1[7] = 0;
  v4i z4 = {0, 0, 0, 0};
  v8i z8 = {0, 0, 0, 0, 0, 0, 0, 0};
  __builtin_amdgcn_tensor_load_to_lds(g0, g1, z4, z4, z8, 0);
}

// ---------------------------------------------------------------------------
// Kernel 1: build logc[NS].  log(conc) for balanced; log(exp(x)) == x for
// unbalanced, so copy log_conc_unbalanced directly.
// ---------------------------------------------------------------------------
__global__ void k_logc(const float* __restrict__ conc_bal,
                       const int* __restrict__ bal_idx,
                       const int* __restrict__ unbal_idx,
                       const float* __restrict__ log_conc_unbal,
                       float* __restrict__ logc) {
  int i = blockIdx.x * blockDim.x + threadIdx.x;
  if (i < NBAL)   logc[bal_idx[i]]   = logf(conc_bal[i]);
  if (i < NUNBAL) logc[unbal_idx[i]] = log_conc_unbal[i];
}

// ---------------------------------------------------------------------------
// Kernel 2: v[r] = exp(log_kcat[r] + sum_s max(-S[s,r],0) * logc[s])
// 256 threads (8 waves); each wave owns a 16-reaction M-tile and reduces over
// all NS species with V_WMMA_F32_16X16X4_F32 (vector in column N=0 of B).
// S tiles (64 species x 128 reactions) stream through LDS via TDM,
// double-buffered: wave 0 issues tensor_load of buffer 1-b while all waves
// consume buffer b.  One s_barrier per tile (issue happens after the barrier
// every wave reaches post-compute on the target buffer).
// ---------------------------------------------------------------------------
#define V_RB 128         // reactions per workgroup
#define V_SB 64          // species rows per tile
#define V_LDST 132       // LDS row stride (floats) = 128 + 4 pad (TDM pad)

__global__ __launch_bounds__(256) void k_flux(
    const float* __restrict__ S,
    const float* __restrict__ logc,
    const float* __restrict__ log_kcat,
    float* __restrict__ v_out) {
  __shared__ __align__(16) float tile[2][V_SB * V_LDST];
  __shared__ float lchunk[2][V_SB];

  const int tid    = threadIdx.x;
  const int lane   = tid & 31;
  const int wave   = tid >> 5;
  const int rbase  = blockIdx.x * V_RB;
  const int m      = lane & 15;          // A/C row M within wave tile
  const int khalf  = (lane >> 4) * 2;    // K pair offset (ISA A layout)
  const int rcol   = wave * 16 + m;      // lane's reaction column in tile
  const bool bsel  = (m == 0);           // B column N==0 holder
  const bool issuer = (tid < 32);        // wave 0 drives the TDM

  v8f acc = {0.f, 0.f, 0.f, 0.f, 0.f, 0.f, 0.f, 0.f};

  if (issuer)
    tdm_load_2d((unsigned)(uintptr_t)&tile[0][0], S + rbase,
                V_RB, V_SB, NR, NS, NR, /*pad 128dw*/6, /*+4dw*/3);

  for (int s0 = 0, it = 0; s0 < NS; s0 += V_SB, ++it) {
    const int buf = it & 1;
    if (issuer) __builtin_amdgcn_s_wait_tensorcnt(0);   // tile[buf] landed
    if (tid < V_SB) lchunk[buf][tid] = logc[s0 + tid];
    __syncthreads();   // tile[buf]/lchunk[buf] visible; all done with tile[1-buf]
    if (issuer && (s0 + V_SB < NS))
      tdm_load_2d((unsigned)(uintptr_t)&tile[buf ^ 1][0],
                  S + (size_t)(s0 + V_SB) * NR + rbase,
                  V_RB, V_SB, NR, NS, NR, 6, 3);

    const float* t = &tile[buf][0];
    const float* lc = &lchunk[buf][0];
#pragma unroll 4
    for (int kk = 0; kk < V_SB; kk += 4) {
      v2f a, b;
      // A[M=m, K] = max(-S[s0+kk+K, rbase+rcol], 0); lane K = khalf, khalf+1
      float x0 = t[(kk + khalf + 0) * V_LDST + rcol];
      float x1 = t[(kk + khalf + 1) * V_LDST + rcol];
      a.x = fmaxf(-x0, 0.f);
      a.y = fmaxf(-x1, 0.f);
      // Unconditional LDS read, then lane select (v_cndmask, no exec dance).
      float t0 = lc[kk + khalf + 0];
      float t1 = lc[kk + khalf + 1];
      b.x = bsel ? t0 : 0.f;
      b.y = bsel ? t1 : 0.f;
      acc = __builtin_amdgcn_wmma_f32_16x16x4_f32(
          false, a, false, b, (short)0, acc, false, false);
    }
  }

  // Column N=0 of C lives in lanes 0 (M=0..7) and 16 (M=8..15), VGPRs 0..7.
  if (m == 0) {
    int mbase = (lane >> 4) * 8;
#pragma unroll
    for (int j = 0; j < 8; ++j) {
      int r = rbase + wave * 16 + mbase + j;
      v_out[r] = expf(log_kcat[r] + acc[j]);
    }
  }
}

// ---------------------------------------------------------------------------
// Kernel 3: dcdt[s] = sum_r S[s,r] * v[r], s in [0, NBAL)
// Wave owns a 16-species M-tile; reduces over all NR reactions.
// Tile = 128 species x 64 reactions, TDM double-buffered as above.
// ---------------------------------------------------------------------------
#define D_SB 128         // species per workgroup
#define D_RB 64          // reactions per tile
#define D_LDST 68        // LDS row stride = 64 + 4 pad

__global__ __launch_bounds__(256) void k_dcdt(
    const float* __restrict__ S,
    const float* __restrict__ v,
    float* __restrict__ dcdt) {
  __shared__ __align__(16) float tile[2][D_SB * D_LDST];
  __shared__ float vchunk[2][D_RB];

  const int tid    = threadIdx.x;
  const int lane   = tid & 31;
  const int wave   = tid >> 5;
  const int sbase  = blockIdx.x * D_SB;
  const int m      = lane & 15;
  const int khalf  = (lane >> 4) * 2;
  const int srow   = wave * 16 + m;      // lane's species row in tile
  const bool bsel  = (m == 0);
  const bool issuer = (tid < 32);

  v8f acc = {0.f, 0.f, 0.f, 0.f, 0.f, 0.f, 0.f, 0.f};

  if (issuer)
    tdm_load_2d((unsigned)(uintptr_t)&tile[0][0], S + (size_t)sbase * NR,
                D_RB, D_SB, NR, NS, NR, /*pad 64dw*/5, /*+4dw*/3);

  for (int r0 = 0, it = 0; r0 < NR; r0 += D_RB, ++it) {
    const int buf = it & 1;
    if (issuer) __builtin_amdgcn_s_wait_tensorcnt(0);
    if (tid < D_RB) vchunk[buf][tid] = v[r0 + tid];
    __syncthreads();
    if (issuer && (r0 + D_RB < NR))
      tdm_load_2d((unsigned)(uintptr_t)&tile[buf ^ 1][0],
                  S + (size_t)sbase * NR + (r0 + D_RB),
                  D_RB, D_SB, NR, NS, NR, 5, 3);

    const float* t = &tile[buf][0];
    const float* vc = &vchunk[buf][0];
#pragma unroll 4
    for (int kk = 0; kk < D_RB; kk += 4) {
      v2f a, b;
      // A[M=m, K] = S[sbase+srow, r0+kk+K]
      a.x = t[srow * D_LDST + kk + khalf + 0];
      a.y = t[srow * D_LDST + kk + khalf + 1];
      float t0 = vc[kk + khalf + 0];
      float t1 = vc[kk + khalf + 1];
      b.x = bsel ? t0 : 0.f;
      b.y = bsel ? t1 : 0.f;
      acc = __builtin_amdgcn_wmma_f32_16x16x4_f32(
          false, a, false, b, (short)0, acc, false, false);
    }
  }

  if (m == 0) {
    int mbase = (lane >> 4) * 8;
#pragma unroll
    for (int j = 0; j < 8; ++j) {
      dcdt[sbase + wave * 16 + mbase + j] = acc[j];
    }
  }
}

// ---------------------------------------------------------------------------
// Launch wrapper.
// Inputs (setup_inputs order):
//   0: conc_balanced [7680] f32       1: S [8192*16384] f32
//   2: balanced_species [7680] i32    3: unbalanced_species [512] i32
//   4: log_conc_unbalanced [512] f32  5: log_kcat [16384] f32
// Output: dcdt [7680] f32.  Workspace: logc (8192 f) + v (16384 f) = 96 KB.
// ---------------------------------------------------------------------------
extern "C" void kernel_launch(void* const* d_in, const int* in_sizes, int n_in,
                              void* d_out, int out_size, void* d_ws, size_t ws_size,
                              hipStream_t stream) {
  const float* conc_bal       = (const float*)d_in[0];
  const float* S              = (const float*)d_in[1];
  const int*   bal_idx        = (const int*)d_in[2];
  const int*   unbal_idx      = (const int*)d_in[3];
  const float* log_conc_unbal = (const float*)d_in[4];
  const float* log_kcat       = (const float*)d_in[5];
  float* out  = (float*)d_out;
  float* logc = (float*)d_ws;          // NS floats
  float* v    = logc + NS;             // NR floats

  k_logc<<<(NS + 255) / 256, 256, 0, stream>>>(conc_bal, bal_idx, unbal_idx,
                                               log_conc_unbal, logc);
  k_flux<<<NR / V_RB, 256, 0, stream>>>(S, logc, log_kcat, v);
  k_dcdt<<<NBAL / D_SB, 256, 0, stream>>>(S, v, out);
}